// CustomConv2d_89653147337370
// MI455X (gfx1250) — compile-verified
//
#include <hip/hip_runtime.h>

// fp32 implicit-GEMM conv using CDNA5 V_WMMA_F32_16X16X4_F32 (wave32).
// M = COUT(128), N = B*H*W(65536), K = CIN*9(1152).
// Per block: 128x128 output tile, 8 waves each computing a 32x64 sub-tile.
// Register-prefetch pipeline hides global (L2) gather latency behind WMMAs.

typedef float v2f __attribute__((ext_vector_type(2)));
typedef float v8f __attribute__((ext_vector_type(8)));

#define BATCH    16
#define CIN_     128
#define COUT_    128
#define HDIM     64
#define WDIM     64
#define PADV     1

#define CC       32      // cin sub-chunk staged per LDS round
#define CCP      34      // LDS row pitch in floats (even -> aligned b64 frag reads)
#define NTILE    128     // output columns per workgroup (2 rows x 64 ow)
#define NTHREADS 256     // 8 wave32 waves
#define NITER    36      // 9 filter taps * 4 cin chunks

__launch_bounds__(NTHREADS)
__global__ void conv2d_wmma_f32_kernel(const float* __restrict__ x,
                                       const float* __restrict__ wgt,
                                       float* __restrict__ out)
{
    __shared__ float As[COUT_ * CCP];   // [m][k]
    __shared__ float Bs[NTILE * CCP];   // [n][k]

    const int tid   = threadIdx.x;
    const int wave  = tid >> 5;
    const int lane  = tid & 31;
    const int lhalf = lane & 15;
    const int koff  = (lane >> 4) << 1;     // 0 for lanes 0-15, 2 for lanes 16-31

    const int blk = blockIdx.x;             // 512 blocks: (batch, output-row-pair)
    const int b   = blk >> 5;
    const int oh0 = (blk & 31) << 1;

    const int mbase = (wave >> 1) << 5;     // 0,32,64,96  : 32 couts per wave
    const int nbase = (wave & 1) << 6;      // 0,64        : 64 cols per wave

    // Per-thread staging coordinates (fixed for whole kernel):
    const int ka    = tid & 31;             // A: fixed k, m = ma0 + 8*i
    const int ma0   = tid >> 5;
    const int nb    = tid & 127;            // B: fixed n, k = kb0 + 2*i
    const int kb0   = tid >> 7;
    const int ohl_b = nb >> 6;
    const int ow_b  = nb & 63;

    float rA[16], rB[16];

    v8f acc[8];                             // [ma][nt] : ma*4+nt
    #pragma unroll
    for (int i = 0; i < 8; ++i)
        acc[i] = (v8f){0.f, 0.f, 0.f, 0.f, 0.f, 0.f, 0.f, 0.f};

    // ---- global -> registers for chunk `iter`
    auto prefetch = [&](int iter) {
        const int kpos = iter >> 2;
        const int c0   = (iter & 3) << 5;
        const int kh   = kpos / 3;
        const int kw   = kpos % 3;
        #pragma unroll
        for (int i = 0; i < 16; ++i) {
            int m = ma0 + (i << 3);
            rA[i] = wgt[(m * CIN_ + (c0 + ka)) * 9 + kpos];
        }
        const int ih = oh0 + ohl_b + kh - PADV;
        const int iw = ow_b + kw - PADV;
        const bool inb = ((unsigned)ih < (unsigned)HDIM) &&
                         ((unsigned)iw < (unsigned)WDIM);
        const float* xp = &x[((b * CIN_ + c0 + kb0) * HDIM + ih) * WDIM + iw];
        #pragma unroll
        for (int i = 0; i < 16; ++i)
            rB[i] = inb ? xp[(size_t)(i << 1) * (HDIM * WDIM)] : 0.0f;
    };

    // ---- registers -> LDS
    auto commit = [&]() {
        #pragma unroll
        for (int i = 0; i < 16; ++i)
            As[(ma0 + (i << 3)) * CCP + ka] = rA[i];
        #pragma unroll
        for (int i = 0; i < 16; ++i)
            Bs[nb * CCP + kb0 + (i << 1)] = rB[i];
    };

    // ---- 8 k-steps (K=4 each) x (2 M-subtiles x 4 N-tiles) WMMAs
    auto compute = [&]() {
        #pragma unroll
        for (int kk = 0; kk < CC; kk += 4) {
            v2f a0 = *(const v2f*)&As[(mbase + lhalf) * CCP + kk + koff];
            v2f a1 = *(const v2f*)&As[(mbase + 16 + lhalf) * CCP + kk + koff];
            #pragma unroll
            for (int nt = 0; nt < 4; ++nt) {
                v2f bf = *(const v2f*)&Bs[(nbase + nt * 16 + lhalf) * CCP + kk + koff];
                acc[nt]     = __builtin_amdgcn_wmma_f32_16x16x4_f32(
                    false, a0, false, bf, (short)0, acc[nt],     false, false);
                acc[4 + nt] = __builtin_amdgcn_wmma_f32_16x16x4_f32(
                    false, a1, false, bf, (short)0, acc[4 + nt], false, false);
            }
        }
    };

    // ---- pipeline: prefetch(i+1) overlaps compute(i); single LDS buffer
    prefetch(0);
    commit();
    __syncthreads();
    for (int iter = 0; iter < NITER - 1; ++iter) {
        prefetch(iter + 1);   // global loads issue now ...
        compute();            // ... and land while 64 WMMAs run
        __syncthreads();      // all waves done reading LDS
        commit();             // s_wait_loadcnt here, after the compute slack
        __syncthreads();
    }
    compute();

    // ---- store. C/D layout: VGPR v holds M = base + v (+8 for lanes 16-31), N = lane%16.
    // Output layout: out[(cout*BATCH + b)*H*W + oh*W + ow]
    const int mrow = mbase + ((lane >> 4) << 3);
    #pragma unroll
    for (int ma = 0; ma < 2; ++ma) {
        #pragma unroll
        for (int nt = 0; nt < 4; ++nt) {
            int ncol = nbase + nt * 16 + lhalf;
            int ohl  = ncol >> 6;
            int ow   = ncol & 63;
            float* op = &out[(((mrow + 16 * ma) * BATCH + b) * HDIM + (oh0 + ohl)) * WDIM + ow];
            #pragma unroll
            for (int v = 0; v < 8; ++v)
                op[(size_t)v * (BATCH * HDIM * WDIM)] = acc[ma * 4 + nt][v];
        }
    }
}

extern "C" void kernel_launch(void* const* d_in, const int* in_sizes, int n_in,
                              void* d_out, int out_size, void* d_ws, size_t ws_size,
                              hipStream_t stream) {
    (void)in_sizes; (void)n_in; (void)d_ws; (void)ws_size; (void)out_size;
    const float* x   = (const float*)d_in[0];
    const float* wgt = (const float*)d_in[1];
    float* out       = (float*)d_out;

    dim3 grid(BATCH * (HDIM / 2));   // 512 workgroups: (b, output-row-pair)
    dim3 block(NTHREADS);            // 8 wave32 waves
    conv2d_wmma_f32_kernel<<<grid, block, 0, stream>>>(x, wgt, out);
}